// LstmRNN_65171833749536
// MI455X (gfx1250) — compile-verified
//
#include <hip/hip_runtime.h>
#include <hip/hip_bf16.h>

// Problem constants (match reference)
#define B_   64
#define T_   128
#define I_   512
#define H_   1024
#define O_   512
#define G_   4096   // 4*H

typedef float v2f __attribute__((ext_vector_type(2)));
typedef float v8f __attribute__((ext_vector_type(8)));

__device__ __forceinline__ float sigmoidf_(float x) {
    return 1.0f / (1.0f + __expf(-x));
}

// ---------------------------------------------------------------------------
// Zero-init kernel (h0, h1, c)
// ---------------------------------------------------------------------------
__global__ void __launch_bounds__(256) zero_f32(float* __restrict__ p, int n) {
    int i = blockIdx.x * 256 + threadIdx.x;
    if (i < n) p[i] = 0.0f;
}

// ---------------------------------------------------------------------------
// Generic fp32 WMMA GEMM + bias:  C[M,N] = A[M,K] @ B[K,N] + bias[N]
// One wave computes a 16x64 strip (4 accumulators share the A fragment).
// Requires M%16==0, N%64==0, K%4==0; grid.x = (M/16)*(N/64)/8 blocks of 256.
// WMMA f32 16x16x4 fragments:
//   A: lane L holds row m=L%16; v0 = A[m][k+2*(L/16)], v1 = A[m][k+2*(L/16)+1]
//   B: lane L holds col n=L%16; v0 = B[k+2*(L/16)][n], v1 = B[k+2*(L/16)+1][n]
//   C/D: lane L holds col n=L%16, rows v + 8*(L/16) for v=0..7
// ---------------------------------------------------------------------------
__global__ void __launch_bounds__(256)
wmma_gemm_bias(const float* __restrict__ A, const float* __restrict__ Bm,
               const float* __restrict__ bias, float* __restrict__ Cout,
               int M, int N, int K)
{
    const int lane = threadIdx.x & 31;
    const int wave = threadIdx.x >> 5;
    const int ngroups = N >> 6;                 // number of 64-col groups
    const int w  = blockIdx.x * 8 + wave;       // global wave id
    const int mt = w / ngroups;
    const int ng = w % ngroups;
    if (mt * 16 >= M) return;
    const int m0 = mt * 16;
    const int n0 = ng * 64;

    const int mrow = lane & 15;                 // A row / B col within tile
    const int kk   = (lane >> 4) * 2;           // K sub-offset for this lane half

    v8f acc0 = {}, acc1 = {}, acc2 = {}, acc3 = {};

    const float* Arow = A + (size_t)(m0 + mrow) * K;
    for (int k = 0; k < K; k += 4) {
        // A fragment: two contiguous floats per lane -> single b64 load
        const v2f a = *(const v2f*)(Arow + k + kk);
        const float* B0 = Bm + (size_t)(k + kk)     * N + n0 + mrow;
        const float* B1 = Bm + (size_t)(k + kk + 1) * N + n0 + mrow;
        v2f b0; b0.x = B0[0];  b0.y = B1[0];
        v2f b1; b1.x = B0[16]; b1.y = B1[16];
        v2f b2; b2.x = B0[32]; b2.y = B1[32];
        v2f b3; b3.x = B0[48]; b3.y = B1[48];
        acc0 = __builtin_amdgcn_wmma_f32_16x16x4_f32(false, a, false, b0, (short)0, acc0, false, false);
        acc1 = __builtin_amdgcn_wmma_f32_16x16x4_f32(false, a, false, b1, (short)0, acc1, false, false);
        acc2 = __builtin_amdgcn_wmma_f32_16x16x4_f32(false, a, false, b2, (short)0, acc2, false, false);
        acc3 = __builtin_amdgcn_wmma_f32_16x16x4_f32(false, a, false, b3, (short)0, acc3, false, false);
    }

    const int half = lane >> 4;
    const int ncol = lane & 15;
    const float bv0 = bias[n0 + ncol];
    const float bv1 = bias[n0 + ncol + 16];
    const float bv2 = bias[n0 + ncol + 32];
    const float bv3 = bias[n0 + ncol + 48];
#pragma unroll
    for (int v = 0; v < 8; ++v) {
        const int mo = m0 + v + 8 * half;
        float* crow = Cout + (size_t)mo * N + n0 + ncol;
        crow[0]  = acc0[v] + bv0;
        crow[16] = acc1[v] + bv1;
        crow[32] = acc2[v] + bv2;
        crow[48] = acc3[v] + bv3;
    }
}

// ---------------------------------------------------------------------------
// Fused LSTM timestep:
//   gates[B,4H] = x_gates[:,t,:] + h_in @ W_hh ; cell update -> h_out, c
// Block = 4 waves (128 threads). Wave g computes the 16x16 tile of gate g at
// (batch-tile mt, h-col-tile nt); all four gate tiles land in LDS, then the
// block applies the LSTM cell pointwise. h double-buffered; c in-place.
// K=1024 is split across FOUR independent accumulator chains (64 WMMAs each)
// for intra-wave ILP -- this kernel is the serial latency path (T=128 steps)
// and each SIMD holds ~1 wave, so WMMA latency must be hidden inside the wave.
// Grid: 4 m-tiles * 64 n-tiles = 256 blocks.
// ---------------------------------------------------------------------------
__global__ void __launch_bounds__(128)
lstm_step(const float* __restrict__ xg,    // [B*T, 4H]
          const float* __restrict__ Whh,   // [H, 4H]
          const float* __restrict__ h_in,  // [B, H]
          float* __restrict__ h_out,       // [B, H]
          float* __restrict__ c,           // [B, H], in-place
          int t)
{
    __shared__ float gtile[4][256];

    const int lane = threadIdx.x & 31;
    const int gate = threadIdx.x >> 5;      // 0..3 = i,f,g,o
    const int mt = blockIdx.x >> 6;         // 4 batch tiles
    const int nt = blockIdx.x & 63;         // 64 h-column tiles
    const int m0 = mt * 16;
    const int n0 = nt * 16;

    const int mrow = lane & 15;
    const int kk   = (lane >> 4) * 2;

    v8f acc0 = {}, acc1 = {}, acc2 = {}, acc3 = {};
    const float* Arow = h_in + (size_t)(m0 + mrow) * H_;
    const float* Bcol = Whh + (size_t)gate * H_ + n0 + mrow;
    for (int k = 0; k < H_; k += 16) {
        const v2f a0 = *(const v2f*)(Arow + k + 0  + kk);
        const v2f a1 = *(const v2f*)(Arow + k + 4  + kk);
        const v2f a2 = *(const v2f*)(Arow + k + 8  + kk);
        const v2f a3 = *(const v2f*)(Arow + k + 12 + kk);
        v2f b0, b1, b2, b3;
        b0.x = Bcol[(size_t)(k + 0  + kk)     * G_];
        b0.y = Bcol[(size_t)(k + 0  + kk + 1) * G_];
        b1.x = Bcol[(size_t)(k + 4  + kk)     * G_];
        b1.y = Bcol[(size_t)(k + 4  + kk + 1) * G_];
        b2.x = Bcol[(size_t)(k + 8  + kk)     * G_];
        b2.y = Bcol[(size_t)(k + 8  + kk + 1) * G_];
        b3.x = Bcol[(size_t)(k + 12 + kk)     * G_];
        b3.y = Bcol[(size_t)(k + 12 + kk + 1) * G_];
        acc0 = __builtin_amdgcn_wmma_f32_16x16x4_f32(false, a0, false, b0, (short)0, acc0, false, false);
        acc1 = __builtin_amdgcn_wmma_f32_16x16x4_f32(false, a1, false, b1, (short)0, acc1, false, false);
        acc2 = __builtin_amdgcn_wmma_f32_16x16x4_f32(false, a2, false, b2, (short)0, acc2, false, false);
        acc3 = __builtin_amdgcn_wmma_f32_16x16x4_f32(false, a3, false, b3, (short)0, acc3, false, false);
    }

    // reduce the four K-chains, add x-gate contribution, stash tile in LDS
    const int half = lane >> 4;
    const int ncol = lane & 15;
#pragma unroll
    for (int v = 0; v < 8; ++v) {
        const int r = v + 8 * half;                     // local row (batch)
        const int bidx = m0 + r;
        const float xv = xg[((size_t)bidx * T_ + t) * G_ + (size_t)gate * H_ + n0 + ncol];
        const float sum = (acc0[v] + acc1[v]) + (acc2[v] + acc3[v]);
        gtile[gate][r * 16 + ncol] = sum + xv;
    }
    __syncthreads();

    // pointwise cell update: 256 elements, 128 threads -> 2 each
    for (int e = threadIdx.x; e < 256; e += 128) {
        const int r  = e >> 4;
        const int cl = e & 15;
        const int bidx = m0 + r;
        const int hcol = n0 + cl;
        const float ii = sigmoidf_(gtile[0][e]);
        const float ff = sigmoidf_(gtile[1][e]);
        const float gv = fmaxf(gtile[2][e], 0.0f);     // relu activation
        const float oo = sigmoidf_(gtile[3][e]);
        const size_t idx = (size_t)bidx * H_ + hcol;
        const float cn = ff * c[idx] + ii * gv;
        c[idx] = cn;
        h_out[idx] = oo * fmaxf(cn, 0.0f);
    }
}

// ---------------------------------------------------------------------------
// Launch: inputs in setup_inputs() order: xs, W_ih, W_hh, b, W_o, b_o
// Workspace layout (floats): xg[B*T*4H] | h0[B*H] | h1[B*H] | c[B*H]
// ---------------------------------------------------------------------------
extern "C" void kernel_launch(void* const* d_in, const int* in_sizes, int n_in,
                              void* d_out, int out_size, void* d_ws, size_t ws_size,
                              hipStream_t stream) {
    const float* xs   = (const float*)d_in[0];
    const float* W_ih = (const float*)d_in[1];
    const float* W_hh = (const float*)d_in[2];
    const float* b    = (const float*)d_in[3];
    const float* W_o  = (const float*)d_in[4];
    const float* b_o  = (const float*)d_in[5];
    float* out = (float*)d_out;

    float* ws = (float*)d_ws;
    float* xg = ws;                                  // 33,554,432 floats
    float* h0 = xg + (size_t)B_ * T_ * G_;
    float* h1 = h0 + (size_t)B_ * H_;
    float* c  = h1 + (size_t)B_ * H_;

    // zero h0, h1, c (contiguous 3*B*H)
    {
        const int n = 3 * B_ * H_;
        zero_f32<<<(n + 255) / 256, 256, 0, stream>>>(h0, n);
    }

    // x_gates = xs @ W_ih + b : M=8192, N=4096, K=512
    {
        const int M = B_ * T_, N = G_, K = I_;
        const int waves = (M / 16) * (N / 64);       // 32768 waves
        wmma_gemm_bias<<<waves / 8, 256, 0, stream>>>(xs, W_ih, b, xg, M, N, K);
    }

    // recurrence: 128 dependent steps, h double-buffered
    for (int t = 0; t < T_; ++t) {
        const float* hin = (t & 1) ? h1 : h0;
        float*      hout = (t & 1) ? h0 : h1;
        lstm_step<<<256, 128, 0, stream>>>(xg, W_hh, hin, hout, c, t);
    }

    // T even -> final h lives in h0. out = h @ W_o + b_o : M=64, N=512, K=1024
    {
        const int M = B_, N = O_, K = H_;
        const int waves = (M / 16) * (N / 64);       // 32 waves
        wmma_gemm_bias<<<waves / 8, 256, 0, stream>>>(h0, W_o, b_o, out, M, N, K);
    }
}